// ObjectDetector_13580686590027
// MI455X (gfx1250) — compile-verified
//
#include <hip/hip_runtime.h>
#include <math.h>

// ---------------------------------------------------------------------------
// Object detector post-processing for MI455X (gfx1250).
// HBM-bound: one 47.2 MB heatmap read ~= 2.0 us at 23.3 TB/s. The heatmap
// fits in the 192 MB L2, so the two refinement passes + collect pass are
// L2-resident. No matmul -> WMMA inapplicable; CDNA5-specific path used is
// the async global->LDS data mover (GLOBAL_LOAD_ASYNC_TO_LDS_B128 +
// s_wait_asynccnt) staging halo tiles for the 3x3 max-pool.
//
// Pipeline:
//   init -> scan(mode0: hist of bits>>16) -> thresh(level0)
//        -> scan(mode1: hist of low16 within crossing bin) -> thresh(level1)
//        -> scan(mode2: collect peaks with bits >= exact 32-bit threshold)
//        -> select (tiny candidate set, LDS-cached, 128 argmax rounds)
// ---------------------------------------------------------------------------

#define HH   384
#define WW   384
#define CC   80
#define KDET 128
#define PEAK_TH 0.1f
#define STRIDE_F 4.0f

#define TY 4
#define TX 32
#define TROWS (TY + 2)
#define TCOLS (TX + 2)
#define ROWF  (TCOLS * CC)        // 2720 floats per LDS halo row
#define TILEF (TROWS * ROWF)      // 16320 floats = 65280 B (< 64 KB static LDS)

#define NBINS 65536
#define CAP   65536
#define SEL_LDS 4096

// workspace layout in 32-bit words (~1.0 MB total)
#define OFF_HIST1   0
#define OFF_HIST2   (NBINS)
#define OFF_CNT     (2 * NBINS)
#define OFF_THR1BIN (2 * NBINS + 1)
#define OFF_ABOVE1  (2 * NBINS + 2)
#define OFF_THR     (2 * NBINS + 3)
#define OFF_CBITS   (2 * NBINS + 8)
#define OFF_CIDX    (OFF_CBITS + CAP)

// ---- CDNA5 async global->LDS copy (per-lane B128) -------------------------
// Builtin signature (from hipcc diagnostic): param0 = v4i __device__* (global,
// non-const), param1 = v4i __shared__* (LDS), then imm offset, imm cpol.
typedef int v4i_vs __attribute__((__vector_size__(16)));
typedef __attribute__((address_space(1))) v4i_vs* gvec128_t;
typedef __attribute__((address_space(3))) v4i_vs* lvec128_t;

__device__ __forceinline__ void async_b128(const float* g, float* l) {
#if __has_builtin(__builtin_amdgcn_global_load_async_to_lds_b128)
  __builtin_amdgcn_global_load_async_to_lds_b128(
      (gvec128_t)g, (lvec128_t)l, 0, 0);
#else
  unsigned loff = (unsigned)(unsigned long long)
      (__attribute__((address_space(3))) void*)l;
  asm volatile("global_load_async_to_lds_b128 %0, %1, off"
               :: "v"(loff), "v"((unsigned long long)g) : "memory");
#endif
}

__device__ __forceinline__ void wait_async0() {
#if __has_builtin(__builtin_amdgcn_s_wait_asynccnt)
  __builtin_amdgcn_s_wait_asynccnt(0);
#else
  asm volatile("s_wait_asynccnt 0" ::: "memory");
#endif
}

// ---------------------------------------------------------------------------
__global__ __launch_bounds__(256)
void det_init_kernel(unsigned* __restrict__ ws) {
  const int n = OFF_CBITS;  // both histograms + counters
  for (int i = blockIdx.x * 256 + threadIdx.x; i < n; i += gridDim.x * 256)
    ws[i] = 0u;
}

// mode 0: histogram of bits>>16; mode 1: histogram of low16 inside crossing
// bin; mode 2: compact candidates with bits >= final threshold.
__global__ __launch_bounds__(256)
void det_scan_kernel(const float* __restrict__ hmap, unsigned* __restrict__ ws,
                     int mode) {
  __shared__ float tile[TILEF];
  const int tid = threadIdx.x;
  const int x0 = blockIdx.x * TX;
  const int y0 = blockIdx.y * TY;
  const int gx_start = x0 - 1;
  const int col_lo = (gx_start < 0) ? 1 : 0;
  const int col_hi = (gx_start + TCOLS > WW) ? (TCOLS - 1) : TCOLS;
  const float NEGINF = -__builtin_inff();

  // Stage halo tile: async B128 copies for in-bounds contiguous spans,
  // LDS -inf fill for SAME-padding halo (matches reduce_window identity).
  for (int r = 0; r < TROWS; ++r) {
    const int gy = y0 - 1 + r;
    float* lrow = &tile[r * ROWF];
    if (gy < 0 || gy >= HH) {
      for (int i = tid; i < ROWF; i += 256) lrow[i] = NEGINF;
    } else {
      for (int i = tid; i < col_lo * CC; i += 256) lrow[i] = NEGINF;
      for (int i = tid; i < (TCOLS - col_hi) * CC; i += 256)
        lrow[col_hi * CC + i] = NEGINF;
      const float* g = hmap + ((size_t)gy * WW + (size_t)(gx_start + col_lo)) * CC;
      float* l = lrow + col_lo * CC;
      const int nvec = ((col_hi - col_lo) * CC) >> 2;  // 16B packets (all aligned)
      for (int i = tid; i < nvec; i += 256) async_b128(g + i * 4, l + i * 4);
    }
  }
  wait_async0();
  __syncthreads();

  unsigned bin1 = 0u, thr = 0u;
  if (mode == 1) bin1 = ws[OFF_THR1BIN];
  if (mode == 2) thr  = ws[OFF_THR];
  unsigned* hist1 = ws + OFF_HIST1;
  unsigned* hist2 = ws + OFF_HIST2;
  unsigned* cnt   = ws + OFF_CNT;
  unsigned* cbits = ws + OFF_CBITS;
  unsigned* cidx  = ws + OFF_CIDX;

  const int elems = TY * TX * CC;  // 10240 per block
  for (int e = tid; e < elems; e += 256) {
    const int ty  = e / (TX * CC);
    const int rem = e - ty * (TX * CC);
    const int tx  = rem / CC;
    const int c   = rem - tx * CC;
    const float* p = &tile[(ty + 1) * ROWF + (tx + 1) * CC + c];
    const float v = p[0];
    if (!(v > PEAK_TH)) continue;
    // peak <=> v is the max of its 3x3 neighborhood (ties pass, like hmax==h)
    const bool pk = v >= p[-CC] && v >= p[CC] &&
                    v >= p[-ROWF] && v >= p[-ROWF - CC] && v >= p[-ROWF + CC] &&
                    v >= p[ ROWF] && v >= p[ ROWF - CC] && v >= p[ ROWF + CC];
    if (!pk) continue;
    const unsigned bits = __float_as_uint(v);  // positive -> uint order == float order
    if (mode == 0) {
      atomicAdd(&hist1[bits >> 16], 1u);
    } else if (mode == 1) {
      if ((bits >> 16) == bin1) atomicAdd(&hist2[bits & 0xFFFFu], 1u);
    } else if (bits >= thr) {
      const unsigned pos = atomicAdd(cnt, 1u);
      if (pos < CAP) {
        cbits[pos] = bits;
        cidx[pos]  = (unsigned)(((y0 + ty) * WW + (x0 + tx)) * CC + c);
      }
    }
  }
}

// Parallel suffix-scan threshold finder: largest bin b with
// above + suffix(b) >= KDET. level 0 -> hist1, writes {bin1, strictly-above};
// level 1 -> hist2, writes final 32-bit threshold (bin1<<16)|bin2.
__global__ __launch_bounds__(1024)
void det_thresh_kernel(unsigned* __restrict__ ws, int level) {
  __shared__ unsigned segsum[1024];
  const unsigned* hist = ws + (level == 0 ? OFF_HIST1 : OFF_HIST2);
  const unsigned above = (level == 0) ? 0u : ws[OFF_ABOVE1];
  const int t = threadIdx.x;
  const int base = t * 64;

  unsigned myseg = 0u;
  for (int i = 0; i < 64; ++i) myseg += hist[base + i];
  segsum[t] = myseg;
  __syncthreads();
  // inclusive suffix scan over 1024 segment sums
  for (int off = 1; off < 1024; off <<= 1) {
    const unsigned a = segsum[t];
    const unsigned b = (t + off < 1024) ? segsum[t + off] : 0u;
    __syncthreads();
    segsum[t] = a + b;
    __syncthreads();
  }
  const unsigned above_seg = above + ((t < 1023) ? segsum[t + 1] : 0u);
  // exactly one thread holds the crossing (if total reaches KDET)
  if (above_seg < KDET && above_seg + myseg >= KDET) {
    unsigned cum = above_seg, bin = 0u, strict = above_seg;
    for (int i = 63; i >= 0; --i) {
      const unsigned h = hist[base + i];
      cum += h;
      if (cum >= KDET) { bin = (unsigned)(base + i); strict = cum - h; break; }
    }
    if (level == 0) {
      ws[OFF_THR1BIN] = bin;
      ws[OFF_ABOVE1]  = strict;
    } else {
      ws[OFF_THR] = (ws[OFF_THR1BIN] << 16) | bin;
    }
  }
  if (t == 0 && above + segsum[0] < KDET) {  // fewer than KDET peaks: take all
    if (level == 0) {
      ws[OFF_THR1BIN] = 0u;
      ws[OFF_ABOVE1]  = above + segsum[0] - hist[0];
    } else {
      ws[OFF_THR] = ws[OFF_THR1BIN] << 16;
    }
  }
}

__global__ __launch_bounds__(1024)
void det_select_kernel(const float* __restrict__ rreg,
                       const float* __restrict__ bbox,
                       unsigned* __restrict__ ws, float* __restrict__ out) {
  __shared__ unsigned lb[SEL_LDS];
  __shared__ unsigned li[SEL_LDS];
  __shared__ unsigned sb_bits[1024];
  __shared__ unsigned sb_idx[1024];
  __shared__ int      sb_pos[1024];
  __shared__ unsigned sel_bits[KDET];
  __shared__ unsigned sel_idx[KDET];
  const int tid = threadIdx.x;
  unsigned n = ws[OFF_CNT];
  if (n > CAP) n = CAP;
  unsigned* cbits = ws + OFF_CBITS;
  unsigned* cidx  = ws + OFF_CIDX;

  // Two-level radix threshold keeps n ~ KDET + ties; cache in LDS when it fits.
  const bool inlds = (n <= SEL_LDS);
  if (inlds) {
    for (unsigned i = tid; i < n; i += 1024) { lb[i] = cbits[i]; li[i] = cidx[i]; }
  }
  __syncthreads();
  unsigned* vb_arr = inlds ? lb : cbits;
  unsigned* vi_arr = inlds ? li : cidx;

  // 128 argmax extractions: (value desc, index asc) == lax.top_k ordering.
  for (int k = 0; k < KDET; ++k) {
    unsigned bbits = 0u, bidx = 0xFFFFFFFFu;
    int bpos = -1;
    for (unsigned i = tid; i < n; i += 1024) {
      const unsigned vb = vb_arr[i];
      if (vb == 0u) continue;  // consumed or empty
      const unsigned vi = vi_arr[i];
      if (vb > bbits || (vb == bbits && vi < bidx)) {
        bbits = vb; bidx = vi; bpos = (int)i;
      }
    }
    sb_bits[tid] = bbits; sb_idx[tid] = bidx; sb_pos[tid] = bpos;
    __syncthreads();
    for (int s = 512; s > 0; s >>= 1) {
      if (tid < s) {
        const unsigned ob = sb_bits[tid + s];
        if (ob > sb_bits[tid] ||
            (ob == sb_bits[tid] && sb_idx[tid + s] < sb_idx[tid])) {
          sb_bits[tid] = ob;
          sb_idx[tid]  = sb_idx[tid + s];
          sb_pos[tid]  = sb_pos[tid + s];
        }
      }
      __syncthreads();
    }
    if (tid == 0) {
      sel_bits[k] = sb_bits[0];
      sel_idx[k]  = sb_idx[0];
      if (sb_pos[0] >= 0) vb_arr[sb_pos[0]] = 0u;  // mark consumed
    }
    __syncthreads();
  }

  // Decode + write outputs: [centroids 128x2 (x,y)][box 128x2][types 128][scores 128]
  if (tid < KDET) {
    const int k = tid;
    const unsigned bits = sel_bits[k];
    unsigned idx = sel_idx[k];
    const bool have = (bits != 0u);
    const float val = have ? __uint_as_float(bits) : -__builtin_inff();
    if (!have) idx = 0u;
    const int c  = (int)(idx % CC);
    const int yx = (int)(idx / CC);
    const int x  = yx % WW;
    const int y  = yx / WW;
    const bool valid = val > PEAK_TH;
    const size_t base = (size_t)y * WW + x;
    const float refx = rreg[base * (2 * CC) + 2 * c];
    const float refy = rreg[base * (2 * CC) + 2 * c + 1];
    const float cx = rintf(((float)x + refx) * STRIDE_F);  // RNE like jnp.round
    const float cy = rintf(((float)y + refy) * STRIDE_F);
    const float b0 = bbox[base * 2 + 0] * STRIDE_F;
    const float b1 = bbox[base * 2 + 1] * STRIDE_F;
    out[2 * k]           = cx;
    out[2 * k + 1]       = cy;
    out[256 + 2 * k]     = b0;
    out[256 + 2 * k + 1] = b1;
    out[512 + k]         = valid ? (float)c : -1.0f;
    out[640 + k]         = valid ? val : 0.0f;
  }
}

// ---------------------------------------------------------------------------
extern "C" void kernel_launch(void* const* d_in, const int* in_sizes, int n_in,
                              void* d_out, int out_size, void* d_ws, size_t ws_size,
                              hipStream_t stream) {
  const float* hmap = (const float*)d_in[0];  // [1,384,384,80]
  const float* rreg = (const float*)d_in[1];  // [1,384,384,160]
  const float* bbox = (const float*)d_in[2];  // [1,384,384,2]
  // d_in[3] = max_dets (always 128; compile-time KDET)
  (void)in_sizes; (void)n_in; (void)out_size; (void)ws_size;
  unsigned* ws = (unsigned*)d_ws;
  float* out = (float*)d_out;

  const dim3 scan_grid(WW / TX, HH / TY);  // 12 x 96 = 1152 workgroups
  det_init_kernel<<<dim3(512), dim3(256), 0, stream>>>(ws);
  det_scan_kernel<<<scan_grid, dim3(256), 0, stream>>>(hmap, ws, 0);
  det_thresh_kernel<<<dim3(1), dim3(1024), 0, stream>>>(ws, 0);
  det_scan_kernel<<<scan_grid, dim3(256), 0, stream>>>(hmap, ws, 1);
  det_thresh_kernel<<<dim3(1), dim3(1024), 0, stream>>>(ws, 1);
  det_scan_kernel<<<scan_grid, dim3(256), 0, stream>>>(hmap, ws, 2);
  det_select_kernel<<<dim3(1), dim3(1024), 0, stream>>>(rreg, bbox, ws, out);
}